// STMambaBlock_214748364959
// MI455X (gfx1250) — compile-verified
//
#include <hip/hip_runtime.h>
#include <math.h>

// ---------------------------------------------------------------------------
// ST-Mamba block for MI455X (gfx1250, wave32, WMMA).
// All GEMMs use V_WMMA_F32_16X16X4_F32 (full fp32 precision; the op is
// bandwidth-bound: ~9.4 GFLOP over ~110MB working set -> L2 resident).
// A-tiles are staged to LDS with global_load_async_to_lds_b128 (ASYNCcnt).
// ---------------------------------------------------------------------------

#define B_   2
#define T_   8
#define HH   32
#define WW   32
#define DIM  128
#define DIN  256
#define NST  16
#define RNK  8
#define HID  512
#define L_   (T_ * HH * WW)   // 8192
#define M_   (B_ * L_)        // 16384

typedef float f32x2 __attribute__((ext_vector_type(2)));
typedef float f32x8 __attribute__((ext_vector_type(8)));

__device__ __forceinline__ float silu_(float x) {
    return x / (1.0f + expf(-x));
}
__device__ __forceinline__ float softplus_(float x) {
    return (x > 20.0f) ? x : log1pf(expf(x));
}

// Byte offset of a shared-memory object inside the wave's LDS allocation.
__device__ __forceinline__ unsigned lds_addr_of(const void* p) {
    return (unsigned)(unsigned long long)
        (const __attribute__((address_space(3))) char*)p;
}

// ---------------------------------------------------------------------------
// [B,T,C,H,W] -> [B, L=T*H*W, C]
// ---------------------------------------------------------------------------
__global__ __launch_bounds__(256) void permute_in_kernel(
    const float* __restrict__ xin, float* __restrict__ x) {
    size_t idx = (size_t)blockIdx.x * 256 + threadIdx.x;   // M_*DIM = 2^21
    int c  = idx & 127;
    int l  = (idx >> 7) & 8191;
    int b  = idx >> 20;
    int t  = l >> 10;
    int hw = l & 1023;
    x[idx] = xin[((((size_t)b * T_ + t) * DIM + c) << 10) + hw];
}

// [B,L,C] -> [B,T,C,H,W]
__global__ __launch_bounds__(256) void permute_out_kernel(
    const float* __restrict__ x, float* __restrict__ xout) {
    size_t idx = (size_t)blockIdx.x * 256 + threadIdx.x;   // 2^21, output layout
    int hw = idx & 1023;
    int c  = (idx >> 10) & 127;
    int t  = (idx >> 17) & 7;
    int b  = idx >> 20;
    xout[idx] = x[((size_t)b * L_ + (size_t)t * 1024 + hw) * DIM + c];
}

// ---------------------------------------------------------------------------
// LayerNorm over last dim (128). One 32-lane wave per row, 8 rows per block.
// ---------------------------------------------------------------------------
__global__ __launch_bounds__(256) void layernorm_kernel(
    const float* __restrict__ x, const float* __restrict__ w,
    const float* __restrict__ b, float* __restrict__ y) {
    int row  = blockIdx.x * 8 + (threadIdx.x >> 5);
    int lane = threadIdx.x & 31;
    const float* xr = x + (size_t)row * DIM;
    float v[4];
    float s = 0.0f;
#pragma unroll
    for (int i = 0; i < 4; ++i) { v[i] = xr[lane + 32 * i]; s += v[i]; }
#pragma unroll
    for (int off = 16; off > 0; off >>= 1) s += __shfl_xor(s, off, 32);
    float mu = s * (1.0f / 128.0f);
    float vs = 0.0f;
#pragma unroll
    for (int i = 0; i < 4; ++i) { float d = v[i] - mu; vs += d * d; }
#pragma unroll
    for (int off = 16; off > 0; off >>= 1) vs += __shfl_xor(vs, off, 32);
    float rstd = rsqrtf(vs * (1.0f / 128.0f) + 1e-5f);
    float* yr = y + (size_t)row * DIM;
#pragma unroll
    for (int i = 0; i < 4; ++i) {
        int c = lane + 32 * i;
        yr[c] = (v[i] - mu) * rstd * w[c] + b[c];
    }
}

// ---------------------------------------------------------------------------
// Generic fp32 WMMA GEMM:  C[M_,N] (+)= A[M_,K] * W[N,K]^T  (+bias) (+act)
// V_WMMA_F32_16X16X4_F32 fragment layout (ISA 7.12.2):
//   A frag (2 VGPR): lane -> row m = lane&15, k = 2*(lane>>4)+j
//   B frag (2 VGPR): lane -> col n = lane&15, k = 2*(lane>>4)+j
//   C/D  (8 VGPR):   lane -> col n = lane&15, row m = i + 8*(lane>>4)
// Block: 256 threads = 8 waves; each wave owns one 16x16 tile. The 16-row
// A tile is staged in LDS once per block via async-to-LDS DMA and shared.
// Out-of-range output columns are computed from a clamped weight row and
// simply never stored (no per-lane load masking -> no EXEC thrash).
// act: 0 = none, 1 = softplus
// ---------------------------------------------------------------------------
template <int K>
__global__ __launch_bounds__(256) void gemm_wmma_f32_kernel(
    const float* __restrict__ A, int lda,
    const float* __restrict__ W,          // [N,K] row-major
    const float* __restrict__ bias,       // [N] or nullptr
    float* __restrict__ C, int ldc,
    int N, int accumulate, int act) {
    constexpr int STR = K + 4;            // pad: bank-conflict-free ds_load_b64
    __shared__ float sA[16 * STR];

    const int row0 = blockIdx.y * 16;

    // ---- async DMA of the 16 x K A-tile into LDS (16B granules) ----
    constexpr int GROUPS = (16 * K) / 4;  // number of 4-float granules
    for (int g = threadIdx.x; g < GROUPS; g += 256) {
        int r = g / (K / 4);
        int k = (g - r * (K / 4)) * 4;
        const float* gp = A + (size_t)(row0 + r) * lda + k;
        unsigned la = lds_addr_of(&sA[r * STR + k]);
        asm volatile("global_load_async_to_lds_b128 %0, %1, off"
                     :: "v"(la), "v"((unsigned long long)(uintptr_t)gp)
                     : "memory");
    }
    asm volatile("s_wait_asynccnt 0x0" ::: "memory");
    __syncthreads();

    const int wave = threadIdx.x >> 5;
    const int lane = threadIdx.x & 31;
    const int col0 = (blockIdx.x * 8 + wave) * 16;
    if (col0 >= N) return;                // wave-uniform exit: EXEC stays all-1s

    const int m    = lane & 15;
    const int half = lane >> 4;           // selects k sub-pair
    const int n    = col0 + m;
    const bool nok = (n < N);
    const int nc   = nok ? n : (N - 1);   // clamped (value unused when !nok)
    const f32x2* W2 = (const f32x2*)(W + (size_t)nc * K);

    f32x8 acc = {0.f, 0.f, 0.f, 0.f, 0.f, 0.f, 0.f, 0.f};
    if (accumulate) {
#pragma unroll
        for (int i = 0; i < 8; ++i)
            acc[i] = C[(size_t)(row0 + i + 8 * half) * ldc + nc];
    }

#pragma unroll
    for (int kk = 0; kk < K; kk += 4) {
        f32x2 a = *(const f32x2*)(&sA[m * STR + kk + 2 * half]);  // ds_load_b64
        f32x2 b = W2[(kk >> 1) + half];                           // global b64
        acc = __builtin_amdgcn_wmma_f32_16x16x4_f32(
            false, a, false, b, (short)0, acc, false, false);
    }

    const float bv = bias ? bias[nc] : 0.0f;
#pragma unroll
    for (int i = 0; i < 8; ++i) {
        float v = acc[i] + bv;
        if (act == 1) v = softplus_(v);
        if (nok) C[(size_t)(row0 + i + 8 * half) * ldc + n] = v;
    }
}

// ---------------------------------------------------------------------------
// Depthwise causal conv1d (width 4) over L + SiLU, direction-aware.
// dir=0: xc[l] = silu(sum_k w[k]*x[l-3+k] + b)   (causal run)
// dir=1: xc[l] = silu(sum_k w[k]*x[l+3-k] + b)   (== flipped causal run)
// x part = first 256 cols of xz[b,l,512].
// ---------------------------------------------------------------------------
__global__ __launch_bounds__(256) void conv_silu_kernel(
    const float* __restrict__ xz, float* __restrict__ xc,
    const float* __restrict__ cw, const float* __restrict__ cb, int dir) {
    size_t idx = (size_t)blockIdx.x * 256 + threadIdx.x;   // M_*DIN = 2^22
    int d = idx & 255;
    int l = (idx >> 8) & 8191;
    int b = idx >> 21;
    const float* base = xz + (size_t)b * L_ * (2 * DIN) + d;
    float acc = cb[d];
#pragma unroll
    for (int k = 0; k < 4; ++k) {
        int off = (dir == 0) ? (l - 3 + k) : (l + 3 - k);
        if (off >= 0 && off < L_)
            acc += cw[d * 4 + k] * base[(size_t)off * (2 * DIN)];
    }
    xc[idx] = silu_(acc);
}

// ---------------------------------------------------------------------------
// Selective-scan. Each 16-lane half-wave owns one channel d; lane n in [0,16)
// owns one state. 8192 independent chains; reduction over N via shfl_xor.
// Fuses y = (scan + xc*D) * silu(z).
// grid = (DIN/16, B_), block = 256 (8 waves -> 16 channels per block).
// ---------------------------------------------------------------------------
__global__ __launch_bounds__(256) void scan_kernel(
    const float* __restrict__ delta, const float* __restrict__ xc,
    const float* __restrict__ xdbl,  const float* __restrict__ xz,
    const float* __restrict__ A_log, const float* __restrict__ Dp,
    float* __restrict__ y, int dir) {
    int lane = threadIdx.x & 31;
    int wave = threadIdx.x >> 5;
    int n    = lane & 15;
    int d    = blockIdx.x * 16 + ((wave << 1) | (lane >> 4));
    int b    = blockIdx.y;

    float a  = -expf(A_log[d * NST + n]);
    float Dd = Dp[d];
    float h  = 0.0f;
    const size_t baseM = (size_t)b * L_;

    for (int t = 0; t < L_; ++t) {
        int l = dir ? (L_ - 1 - t) : t;
        size_t rowM = baseM + l;
        int ln = dir ? (l - 1) : (l + 1);
        if (ln >= 0 && ln < L_) {   // hide next step's load latency
            __builtin_prefetch(&delta[(baseM + ln) * DIN + d], 0, 1);
            __builtin_prefetch(&xdbl[(baseM + ln) * 40 + 8 + n], 0, 1);
        }
        float dl  = delta[rowM * DIN + d];
        float xcv = xc[rowM * DIN + d];
        float Bv  = xdbl[rowM * 40 + RNK + n];
        float Cv  = xdbl[rowM * 40 + RNK + NST + n];
        float dA  = expf(dl * a);
        h = dA * h + (dl * xcv) * Bv;
        float yv = h * Cv;
        yv += __shfl_xor(yv, 1, 32);
        yv += __shfl_xor(yv, 2, 32);
        yv += __shfl_xor(yv, 4, 32);
        yv += __shfl_xor(yv, 8, 32);
        if (n == 0) {
            float zv = xz[rowM * (2 * DIN) + DIN + d];
            y[rowM * DIN + d] = (yv + xcv * Dd) * silu_(zv);
        }
    }
}

// ---------------------------------------------------------------------------
__global__ __launch_bounds__(256) void add_kernel(
    float* __restrict__ a, const float* __restrict__ b, size_t nelem) {
    size_t i = (size_t)blockIdx.x * 256 + threadIdx.x;
    if (i < nelem) a[i] += b[i];
}

// ---------------------------------------------------------------------------
// 3x3 depthwise conv (SAME) over [B*T, 32, 32, 512] + bias + exact GELU.
// ---------------------------------------------------------------------------
__global__ __launch_bounds__(256) void dwconv_gelu_kernel(
    const float* __restrict__ h, const float* __restrict__ w,
    const float* __restrict__ bias, float* __restrict__ outp) {
    size_t idx = (size_t)blockIdx.x * 256 + threadIdx.x;   // M_*HID = 2^23
    int c   = idx & 511;
    int xq  = (idx >> 9) & 31;
    int yq  = (idx >> 14) & 31;
    int img = idx >> 19;
    float acc = bias[c];
#pragma unroll
    for (int dy = 0; dy < 3; ++dy) {
        int yy = yq + dy - 1;
        if (yy < 0 || yy > 31) continue;
#pragma unroll
        for (int dx = 0; dx < 3; ++dx) {
            int xx = xq + dx - 1;
            if (xx < 0 || xx > 31) continue;
            acc += w[(dy * 3 + dx) * HID + c] *
                   h[(((size_t)img * 32 + yy) * 32 + xx) * HID + c];
        }
    }
    outp[idx] = 0.5f * acc * (1.0f + erff(acc * 0.70710678118654752f));
}

// ---------------------------------------------------------------------------
template <int K>
static void launch_gemm(const float* A, int lda, const float* W,
                        const float* bias, float* C, int ldc, int N,
                        int accumulate, int act, hipStream_t stream) {
    dim3 grid((N + 127) / 128, M_ / 16);
    gemm_wmma_f32_kernel<K><<<grid, 256, 0, stream>>>(A, lda, W, bias, C, ldc,
                                                      N, accumulate, act);
}

extern "C" void kernel_launch(void* const* d_in, const int* in_sizes, int n_in,
                              void* d_out, int out_size, void* d_ws, size_t ws_size,
                              hipStream_t stream) {
    const float* x_in       = (const float*)d_in[0];
    const float* norm1_w    = (const float*)d_in[1];
    const float* norm1_b    = (const float*)d_in[2];
    const float* in_proj_w  = (const float*)d_in[3];   // [512,128]
    const float* conv_w     = (const float*)d_in[4];   // [256,1,4]
    const float* conv_b     = (const float*)d_in[5];
    const float* x_proj_w   = (const float*)d_in[6];   // [40,256]
    const float* dt_proj_w  = (const float*)d_in[7];   // [256,8]
    const float* dt_proj_b  = (const float*)d_in[8];
    const float* A_log      = (const float*)d_in[9];   // [256,16]
    const float* Dp         = (const float*)d_in[10];
    const float* out_proj_w = (const float*)d_in[11];  // [128,256]
    const float* norm2_w    = (const float*)d_in[12];
    const float* norm2_b    = (const float*)d_in[13];
    const float* fc1_w      = (const float*)d_in[14];  // [512,128]
    const float* fc1_b      = (const float*)d_in[15];
    const float* dw_w       = (const float*)d_in[16];  // [3,3,1,512]
    const float* dw_b       = (const float*)d_in[17];
    const float* fc2_w      = (const float*)d_in[18];  // [128,512]
    const float* fc2_b      = (const float*)d_in[19];
    float* out = (float*)d_out;

    // Workspace pool (floats): total ~106.5 MiB
    float* pool  = (float*)d_ws;
    float* xbuf  = pool;                              // [M,128] shortcut / x2 / final
    float* xn    = xbuf  + (size_t)M_ * DIM;          // [M,128] LN out (reused)
    float* xz    = xn    + (size_t)M_ * DIM;          // [M,512] in_proj out; later MLP hidden
    float* outb  = xz    + (size_t)M_ * 2 * DIN;      // [M,128] mamba out accum
    float* xdbl  = outb  + (size_t)M_ * DIM;          // [M,40]
    float* xc    = xdbl  + (size_t)M_ * 40;           // [M,256]
    float* delta = xc    + (size_t)M_ * DIN;          // [M,256]
    float* yb    = delta + (size_t)M_ * DIN;          // [M,256]
    float* h2    = xc;                                // [M,512] reuses xc+delta in MLP phase

    permute_in_kernel<<<(M_ * DIM) / 256, 256, 0, stream>>>(x_in, xbuf);
    layernorm_kernel<<<M_ / 8, 256, 0, stream>>>(xbuf, norm1_w, norm1_b, xn);
    // xz = xn @ in_proj_w^T        [M,512]
    launch_gemm<DIM>(xn, DIM, in_proj_w, nullptr, xz, 2 * DIN, 2 * DIN, 0, 0, stream);

    for (int dir = 0; dir < 2; ++dir) {
        conv_silu_kernel<<<(M_ * DIN) / 256, 256, 0, stream>>>(xz, xc, conv_w, conv_b, dir);
        // x_dbl = xc @ x_proj_w^T   [M,40]
        launch_gemm<DIN>(xc, DIN, x_proj_w, nullptr, xdbl, 40, 40, 0, 0, stream);
        // delta = softplus(dt @ dt_proj_w^T + b)   [M,256] (dt = x_dbl[:, :8])
        launch_gemm<RNK>(xdbl, 40, dt_proj_w, dt_proj_b, delta, DIN, DIN, 0, 1, stream);
        scan_kernel<<<dim3(DIN / 16, B_), 256, 0, stream>>>(delta, xc, xdbl, xz,
                                                            A_log, Dp, yb, dir);
        // outb (+)= y @ out_proj_w^T   [M,128]
        launch_gemm<DIN>(yb, DIN, out_proj_w, nullptr, outb, DIM, DIM, dir, 0, stream);
    }

    add_kernel<<<(M_ * DIM) / 256, 256, 0, stream>>>(xbuf, outb, (size_t)M_ * DIM);
    layernorm_kernel<<<M_ / 8, 256, 0, stream>>>(xbuf, norm2_w, norm2_b, xn);
    // h = xn @ fc1_w^T + b          [M,512]
    launch_gemm<DIM>(xn, DIM, fc1_w, fc1_b, xz, HID, HID, 0, 0, stream);
    dwconv_gelu_kernel<<<(M_ * HID) / 256, 256, 0, stream>>>(xz, dw_w, dw_b, h2);
    // xbuf += h2 @ fc2_w^T + b      [M,128]
    launch_gemm<HID>(h2, HID, fc2_w, fc2_b, xbuf, DIM, DIM, 1, 0, stream);
    permute_out_kernel<<<(M_ * DIM) / 256, 256, 0, stream>>>(xbuf, out);
}